// Decoder_6064493822277
// MI455X (gfx1250) — compile-verified
//
#include <hip/hip_runtime.h>

// ---------------------------------------------------------------------------
// 2-layer GCN for MI455X (gfx1250).
//   GEMMs:  V_WMMA_F32_16X16X4_F32 (native fp32 matrix path, full precision)
//   Aggreg: coalesced per-wave gathers + GLOBAL_ATOMIC_ADD_F32 scatter.
// Working set (~130 MB) is L2-resident (192 MB), so the aggregation kernels
// are written to maximize L2 throughput: one wave per edge, lane = feature.
// ---------------------------------------------------------------------------

typedef __attribute__((ext_vector_type(2))) float v2f;
typedef __attribute__((ext_vector_type(8))) float v8f;

__device__ __forceinline__ void atomAddF(float* p, float v) {
    // hardware GLOBAL_ATOMIC_ADD_F32 (no CAS loop)
    unsafeAtomicAdd(p, v);
}

// deg[i] = 1.0 accounts for the self-loop edge weight (PyG gcn_norm default).
__global__ __launch_bounds__(256) void k_init_deg(float* __restrict__ deg, int n) {
    int i = blockIdx.x * blockDim.x + threadIdx.x;
    if (i < n) deg[i] = 1.0f;
}

__global__ __launch_bounds__(256) void k_scatter_deg(const long long* __restrict__ col,
                                                     const float* __restrict__ ew,
                                                     float* __restrict__ deg, int E) {
    int e = blockIdx.x * blockDim.x + threadIdx.x;
    if (e < E) atomAddF(&deg[(int)col[e]], ew[e]);
}

__global__ __launch_bounds__(256) void k_dinv(const float* __restrict__ deg,
                                              float* __restrict__ dinv, int n) {
    int i = blockIdx.x * blockDim.x + threadIdx.x;
    if (i < n) {
        float d = deg[i];
        dinv[i] = d > 0.0f ? rsqrtf(d) : 0.0f;
    }
}

// Y[n x FOUT] = X[n x FIN] @ W[FIN x FOUT], one wave per 16x16 tile.
// fp32 WMMA, K advances by 4 per v_wmma_f32_16x16x4_f32.
// A 16x4 layout: lanes 0-15 -> K={0,1}, lanes 16-31 -> K={2,3} (ISA 7.12.2).
// B 4x16 layout: row striped across lanes within a VGPR, same K halving.
// C/D: VGPR j holds row M = j + 8*(lane>=16), col N = lane&15.
template <int FIN, int FOUT>
__global__ __launch_bounds__(256) void k_gemm_wmma(const float* __restrict__ X,
                                                   const float* __restrict__ W,
                                                   float* __restrict__ Y, int n) {
    const int wave = (blockIdx.x * blockDim.x + threadIdx.x) >> 5;
    const int lane = threadIdx.x & 31;
    constexpr int TN = FOUT / 16;
    const int tiles = (n / 16) * TN;
    if (wave >= tiles) return;                 // whole wave exits: EXEC stays all-1s
    const int tm  = wave / TN;
    const int tn  = wave % TN;
    const int hi  = lane >> 4;                 // 0: lanes 0-15, 1: lanes 16-31
    const int l15 = lane & 15;

    const float* xr = X + (size_t)(tm * 16 + l15) * FIN;   // A row for this lane
    const float* wc = W + tn * 16 + l15;                   // B column for this lane

    v8f acc = {};
#pragma unroll
    for (int k0 = 0; k0 < FIN; k0 += 4) {
        v2f a, b;
        a.x = xr[k0 + 2 * hi + 0];
        a.y = xr[k0 + 2 * hi + 1];
        b.x = wc[(size_t)(k0 + 2 * hi + 0) * FOUT];
        b.y = wc[(size_t)(k0 + 2 * hi + 1) * FOUT];
        // 8 args: (neg_a, A, neg_b, B, c_mod, C, reuse_a, reuse_b)
        acc = __builtin_amdgcn_wmma_f32_16x16x4_f32(false, a, false, b,
                                                    (short)0, acc, false, false);
    }

    float* yr = Y + tn * 16 + l15;
#pragma unroll
    for (int j = 0; j < 8; ++j) {
        yr[(size_t)(tm * 16 + 8 * hi + j) * FOUT] = acc[j];
    }
}

// Self-loop contribution + optional bias:  Y[i,f] = dinv[i]^2 * X[i,f] (+ bias[f])
template <int F>
__global__ __launch_bounds__(256) void k_self_init(const float* __restrict__ dinv,
                                                   const float* __restrict__ X,
                                                   const float* __restrict__ bias,
                                                   float* __restrict__ Y, int n) {
    int t = blockIdx.x * blockDim.x + threadIdx.x;
    if (t >= n * F) return;
    int   i  = t / F;
    float di = dinv[i];
    float v  = di * di * X[t];
    if (bias) v += bias[t & (F - 1)];
    Y[t] = v;
}

// Edge scatter: one wave per edge, lane = feature (strided by 32 for F=64).
// Gathers X[row,:] coalesced, atomically accumulates into OUT[col,:].
template <int F>
__global__ __launch_bounds__(256) void k_edge_agg(const long long* __restrict__ row,
                                                  const long long* __restrict__ col,
                                                  const float* __restrict__ ew,
                                                  const float* __restrict__ dinv,
                                                  const float* __restrict__ X,
                                                  float* __restrict__ OUT, int E) {
    int t    = blockIdx.x * blockDim.x + threadIdx.x;
    int e    = t >> 5;
    int lane = t & 31;
    if (e >= E) return;
    int   r   = (int)row[e];
    int   c   = (int)col[e];
    float nrm = dinv[r] * ew[e] * dinv[c];
#pragma unroll
    for (int f = lane; f < F; f += 32) {
        atomAddF(&OUT[(size_t)c * F + f], X[(size_t)r * F + f] * nrm);
    }
}

// Bias + ReLU after aggregation completes (layer 1 only).
template <int F>
__global__ __launch_bounds__(256) void k_bias_relu(float* __restrict__ Y,
                                                   const float* __restrict__ bias, int n) {
    int t = blockIdx.x * blockDim.x + threadIdx.x;
    if (t >= n * F) return;
    float v = Y[t] + bias[t & (F - 1)];
    Y[t] = v > 0.0f ? v : 0.0f;
}

static inline int nblk(long long total, int b) { return (int)((total + b - 1) / b); }

extern "C" void kernel_launch(void* const* d_in, const int* in_sizes, int n_in,
                              void* d_out, int out_size, void* d_ws, size_t ws_size,
                              hipStream_t stream) {
    const float*     z  = (const float*)d_in[0];      // [n,64]
    const long long* ei = (const long long*)d_in[1];  // [2,E] int64
    const float*     ew = (const float*)d_in[2];      // [E]
    const float*     W1 = (const float*)d_in[3];      // [64,64]
    const float*     b1 = (const float*)d_in[4];      // [64]
    const float*     W2 = (const float*)d_in[5];      // [64,32]
    const float*     b2 = (const float*)d_in[6];      // [32]
    float*           out = (float*)d_out;             // [n,32]

    const int E = in_sizes[2];
    const int n = in_sizes[0] / 64;                   // latent = 64 (n % 16 == 0 here)
    const long long* row = ei;
    const long long* col = ei + E;

    // workspace layout (floats): deg | dinv | x1[n*64] | h1[n*64] | x2[n*32]
    float* ws   = (float*)d_ws;
    float* deg  = ws;
    float* dinv = deg + n;
    float* x1   = dinv + n;
    float* h1   = x1 + (size_t)n * 64;
    float* x2   = h1 + (size_t)n * 64;

    const int B = 256;

    // --- normalization: deg = 1 + segment_sum(ew, col);  dinv = rsqrt(deg) ---
    k_init_deg<<<nblk(n, B), B, 0, stream>>>(deg, n);
    k_scatter_deg<<<nblk(E, B), B, 0, stream>>>(col, ew, deg, E);
    k_dinv<<<nblk(n, B), B, 0, stream>>>(deg, dinv, n);

    // --- layer 1 ---
    {
        long long waves = (long long)(n / 16) * (64 / 16);
        k_gemm_wmma<64, 64><<<nblk(waves * 32, B), B, 0, stream>>>(z, W1, x1, n);
    }
    k_self_init<64><<<nblk((long long)n * 64, B), B, 0, stream>>>(dinv, x1, nullptr, h1, n);
    k_edge_agg<64><<<nblk((long long)E * 32, B), B, 0, stream>>>(row, col, ew, dinv, x1, h1, E);
    k_bias_relu<64><<<nblk((long long)n * 64, B), B, 0, stream>>>(h1, b1, n);

    // --- layer 2 ---
    {
        long long waves = (long long)(n / 16) * (32 / 16);
        k_gemm_wmma<64, 32><<<nblk(waves * 32, B), B, 0, stream>>>(h1, W2, x2, n);
    }
    k_self_init<32><<<nblk((long long)n * 32, B), B, 0, stream>>>(dinv, x2, b2, out, n);
    k_edge_agg<32><<<nblk((long long)E * 32, B), B, 0, stream>>>(row, col, ew, dinv, x2, out, E);
}